// ODEFunction_56083682951493
// MI455X (gfx1250) — compile-verified
//
#include <hip/hip_runtime.h>

// ---------------------------------------------------------------------------
// COO SpMM  out[dst] += w * x[src]  (E=3.2M, D=64)  + nan/inf sanitize + clamp
//
// MI455X reasoning: 0.41 GFLOP vs ~1.7 GB of L2-resident traffic (x = 25.6MB,
// out = 25.6MB both fit in 192MB L2) -> purely L2-gather/atomic bound.
// Structure: zero(out) -> wave-tiled gather/scale/scatter -> epilogue.
// diag(w) * X_gathered for each 16-edge x 16-feature tile runs on the matrix
// pipe via chained V_WMMA_F32_16X16X4_F32 (full f32 precision); the D layout
// (lane = feature col, VGPR = edge row) feeds per-lane global_atomic_add_f32.
//
// Round-1 fixes: (a) scatter guard removed -- tail edges have w=0 => c[v]==0
// and dst==0, so an unconditional atomicAdd of 0.0 is a no-op; this kills the
// s_and_saveexec/s_cbranch_execz per atomic. (b) src/dst shuffles and the
// diag-A selects hoisted out of the 4x feature loop (ds 64->16 per tile).
// ---------------------------------------------------------------------------

typedef __attribute__((ext_vector_type(2))) float v2f;
typedef __attribute__((ext_vector_type(8))) float v8f;

#define D_FEAT   64
#define CLAMP_V  20.0f

// ---------------------------------------------------------------------------
__global__ void spmm_zero_kernel(float* __restrict__ out, int n) {
  int i = blockIdx.x * blockDim.x + threadIdx.x;
  int stride = gridDim.x * blockDim.x;
  for (; i < n; i += stride) out[i] = 0.0f;
}

// ---------------------------------------------------------------------------
// One wave32 handles a tile of 16 edges x 64 features.
// For each 16-feature slab:  D(16x16) = sum_{k=0..3} A_k(16x4) * B_k(4x16)
//   A_k = K-slice of diag(w[tile edges])     (lane = M, half*2 VGPRs = K)
//   B_k = gathered x rows for edges 4k..4k+3 (lane = N, half*2 VGPRs = K)
// Then lane n scatters D[v + 8*half][n] into out[dst[row]*64 + f + n].
// No divergent control flow anywhere: EXEC stays all-1s for every WMMA.
__global__ __launch_bounds__(256)
void spmm_wmma_scatter_kernel(const float* __restrict__ x,
                              const int*   __restrict__ edge_src,
                              const int*   __restrict__ edge_dst,
                              const float* __restrict__ edge_w,
                              float*       __restrict__ out,
                              int n_edges) {
  const int  lane = threadIdx.x & 31;
  const int  wave = threadIdx.x >> 5;           // 8 waves / block
  const int  m    = lane & 15;                  // edge row (A) / feature col (B,D)
  const int  half = lane >> 4;                  // K-pair selector
  const long tile = (long)blockIdx.x * 8 + wave;
  const long e    = tile * 16 + m;

  const bool  ev    = e < (long)n_edges;
  const float w_e   = ev ? edge_w[e]   : 0.0f;  // w=0 zeroes the A row for tail
  const int   src_e = ev ? edge_src[e] : 0;
  const int   dst_e = ev ? edge_dst[e] : 0;     // 0 => in-bounds dummy target

  // ---- hoisted, feature-tile-invariant state -----------------------------
  // diag(w) A-slices: a0[k] = A[m, 4k+2*half], a1[k] = A[m, 4k+2*half+1]
  float a0[4], a1[4];
  // gathered source rows for this lane's K pair in each slice
  const float* brow0[4];
  const float* brow1[4];
#pragma unroll
  for (int k = 0; k < 4; ++k) {
    const int kb = k * 4 + half * 2;
    a0[k] = (m == kb)     ? w_e : 0.0f;
    a1[k] = (m == kb + 1) ? w_e : 0.0f;
    const int s0 = __shfl(src_e, kb,     32);
    const int s1 = __shfl(src_e, kb + 1, 32);
    brow0[k] = x + (size_t)s0 * D_FEAT + m;
    brow1[k] = x + (size_t)s1 * D_FEAT + m;
  }
  // destination offsets: lane n handles rows v + 8*half; out index < 6.4M -> u32
  unsigned dofs[8];
#pragma unroll
  for (int v = 0; v < 8; ++v) {
    const int d = __shfl(dst_e, v + half * 8, 32);
    dofs[v] = (unsigned)d * D_FEAT + m;
  }

  // ---- 4 feature slabs: gather -> WMMA chain -> atomic scatter -----------
#pragma unroll
  for (int ft = 0; ft < 4; ++ft) {
    const int f = ft * 16;
    v8f c = {0.f, 0.f, 0.f, 0.f, 0.f, 0.f, 0.f, 0.f};

#pragma unroll
    for (int k = 0; k < 4; ++k) {
      v2f a; a.x = a0[k]; a.y = a1[k];
      v2f b; b.x = brow0[k][f]; b.y = brow1[k][f];
      c = __builtin_amdgcn_wmma_f32_16x16x4_f32(
              /*neg_a=*/false, a, /*neg_b=*/false, b,
              /*c_mod=*/(short)0, c, /*reuse_a=*/false, /*reuse_b=*/false);
    }

    // Unconditional scatter: tail rows contribute exactly 0.0 to out[0+..].
#pragma unroll
    for (int v = 0; v < 8; ++v) {
      (void)__hip_atomic_fetch_add(out + dofs[v] + f, c[v],
                                   __ATOMIC_RELAXED, __HIP_MEMORY_SCOPE_AGENT);
    }
  }
}

// ---------------------------------------------------------------------------
// nan_to_num(nan=0, +inf=1e6, -inf=-1e6) then clip(+-20)  ==  NaN->0, clamp.
__global__ void spmm_epilogue_kernel(float* __restrict__ out, int n) {
  int i = blockIdx.x * blockDim.x + threadIdx.x;
  int stride = gridDim.x * blockDim.x;
  for (; i < n; i += stride) {
    float v = out[i];
    v = (v != v) ? 0.0f : v;                       // NaN -> 0
    v = fminf(CLAMP_V, fmaxf(-CLAMP_V, v));        // +/-inf -> 1e6 -> +/-20
    out[i] = v;
  }
}

// ---------------------------------------------------------------------------
extern "C" void kernel_launch(void* const* d_in, const int* in_sizes, int n_in,
                              void* d_out, int out_size, void* d_ws, size_t ws_size,
                              hipStream_t stream) {
  // inputs: t, x, edge_src, edge_dst, edge_w
  const float* x        = (const float*)d_in[1];
  const int*   edge_src = (const int*)  d_in[2];
  const int*   edge_dst = (const int*)  d_in[3];
  const float* edge_w   = (const float*)d_in[4];
  float*       out      = (float*)d_out;
  const int    E        = in_sizes[2];

  // 1) zero the accumulator (d_out is poisoned by the harness)
  {
    int blocks = (out_size + 1023) / 1024;
    if (blocks > 32768) blocks = 32768;
    if (blocks < 1) blocks = 1;
    spmm_zero_kernel<<<blocks, 1024, 0, stream>>>(out, out_size);
  }

  // 2) WMMA-scaled gather/scatter: 16 edges per wave, 8 waves per block
  {
    long tiles  = ((long)E + 15) / 16;
    long blocks = (tiles + 7) / 8;               // E=3.2M -> 25,000 blocks exactly
    spmm_wmma_scatter_kernel<<<(int)blocks, 256, 0, stream>>>(
        x, edge_src, edge_dst, edge_w, out, E);
  }

  // 3) sanitize + clamp epilogue
  {
    int blocks = (out_size + 1023) / 1024;
    if (blocks > 32768) blocks = 32768;
    if (blocks < 1) blocks = 1;
    spmm_epilogue_kernel<<<blocks, 1024, 0, stream>>>(out, out_size);
  }
}